// MBGCNUD_67954972557828
// MI455X (gfx1250) — compile-verified
//
#include <hip/hip_runtime.h>
#include <hip/hip_bf16.h>

// MBGCN forward for MI455X (gfx1250, wave32, WMMA f32 16x16x4).
// All big GEMMs run on the matrix pipe via V_WMMA_F32_16X16X4_F32 (exact fp32).
// The K-head attention (logits -> softmax over heads -> *rel -> weighted sum)
// is fused (256MB r tensor never materialized); one wave per head per 16-user
// strip to keep VGPR pressure ~60 (no spills), softmax coupled through LDS.

#define UDIM 4096
#define IDIM 4096
#define EDIM 128
#define KH   4
#define DH   32
#define RREL 3
#define BB   512
#define LLEN 100
#define LAMB_C   0.5f
#define L2NORM_C 1e-4f
#define EPS_C    1e-8f

typedef float v2f __attribute__((ext_vector_type(2)));
typedef float v8f __attribute__((ext_vector_type(8)));

__device__ __forceinline__ v8f wmma_f32(v2f a, v2f b, v8f c) {
  // D = A(16x4) * B(4x16) + C, fp32 exact
  return __builtin_amdgcn_wmma_f32_16x16x4_f32(false, a, false, b, (short)0, c, false, false);
}

// ---------------------------------------------------------------- utility ---
__global__ void zero_kernel(float* __restrict__ p, int n) {
  int i = blockIdx.x * blockDim.x + threadIdx.x;
  if (i < n) p[i] = 0.0f;
}

// ubw[u,r] = deg[u,r]*w[r] / (sum_r deg[u,r]*w[r] + EPS)
__global__ void ubw_kernel(const float* __restrict__ deg, const float* __restrict__ w,
                           float* __restrict__ ubw) {
  int u = blockIdx.x * blockDim.x + threadIdx.x;
  if (u >= UDIM) return;
  float t = 0.f;
#pragma unroll
  for (int r = 0; r < RREL; ++r) t += deg[u * RREL + r] * w[r];
  float inv = 1.0f / (t + EPS_C);
#pragma unroll
  for (int r = 0; r < RREL; ++r) ubw[u * RREL + r] = deg[u * RREL + r] * w[r] * inv;
}

// out[m, :] = in[m, :] / (deg[m*degStride] + EPS), in place allowed
__global__ void row_scale_kernel(float* __restrict__ data, const float* __restrict__ deg,
                                 int degStride, int n, int ncols) {
  int i = blockIdx.x * blockDim.x + threadIdx.x;
  if (i >= n) return;
  int m = i / ncols;
  data[i] = data[i] / (deg[(size_t)m * degStride] + EPS_C);
}

// ------------------------------------------------- head projection + l2norm --
// val[d] = relu(sum_e W_K[k,d,e]*emb[row,e] + B_K[k,d]); l2norm over d (32 lanes)
// Item mode writes two PAIR-INTERLEAVED layouts for 8-byte WMMA B-fragment loads:
//   p_dI: element (k,d,i) at ((k*D/2 + d/2)*I + i)*2 + (d&1)   (logits B)
//   p_iD: element (k,i,d) at ((k*I/2 + i/2)*D + d)*2 + (i&1)   (weighted-sum B)
// User mode writes s_u0 as [U,K,D].
__global__ void head_proj_kernel(const float* __restrict__ emb,
                                 const float* __restrict__ W_K,
                                 const float* __restrict__ B_K,
                                 int Nrows,
                                 float* __restrict__ out_p_dI,
                                 float* __restrict__ out_p_iD,
                                 float* __restrict__ out_ukd) {
  int bid = blockIdx.x;
  int k = bid & 3;
  int row = bid >> 2;
  int d = threadIdx.x;  // 0..31
  const float4* wrow = (const float4*)(W_K + (size_t)k * DH * EDIM + (size_t)d * EDIM);
  const float4* er = (const float4*)(emb + (size_t)row * EDIM);
  float s = B_K[k * DH + d];
#pragma unroll 8
  for (int e = 0; e < EDIM / 4; ++e) {
    float4 a = wrow[e], b = er[e];
    s += a.x * b.x + a.y * b.y + a.z * b.z + a.w * b.w;
  }
  s = fmaxf(s, 0.0f);
  float sq = s * s;
#pragma unroll
  for (int off = 16; off > 0; off >>= 1) sq += __shfl_xor(sq, off, 32);
  float v = s / fmaxf(sqrtf(sq), 1e-12f);
  if (out_p_dI)
    out_p_dI[((size_t)(k * (DH / 2) + (d >> 1)) * Nrows + row) * 2 + (d & 1)] = v;
  if (out_p_iD)
    out_p_iD[((size_t)(k * (Nrows / 2) + (row >> 1)) * DH + d) * 2 + (row & 1)] = v;
  if (out_ukd) out_ukd[(size_t)row * KH * DH + k * DH + d] = v;
}

// ----------------------------------------------------------- WMMA fp32 GEMM --
// C[M,N] = A[M,Kd] * B[Kd,N], row-major; TRANSA: A element (m,k) = A[k*lda+m].
// One wave (32 threads) per 16x16 C tile. M,N mult of 16, Kd mult of 4.
template <bool TRANSA>
__global__ void wmma_gemm_f32(const float* __restrict__ A, const float* __restrict__ B,
                              float* __restrict__ C, int M, int N, int Kd,
                              int lda, int ldb, int ldc) {
  const int lane = threadIdx.x;
  const int half = lane >> 4;   // 0 or 1
  const int mn = lane & 15;
  const int nBase = blockIdx.x * 16;
  const int mBase = blockIdx.y * 16;
  v8f acc = {0.f, 0.f, 0.f, 0.f, 0.f, 0.f, 0.f, 0.f};
#pragma unroll 2
  for (int k = 0; k < Kd; k += 4) {
    int ka = k + half * 2;
    v2f a, b;
    if (TRANSA) {
      a.x = A[(size_t)ka * lda + mBase + mn];
      a.y = A[(size_t)(ka + 1) * lda + mBase + mn];
    } else {
      a = *(const v2f*)(A + (size_t)(mBase + mn) * lda + ka);
    }
    b.x = B[(size_t)ka * ldb + nBase + mn];
    b.y = B[(size_t)(ka + 1) * ldb + nBase + mn];
    acc = wmma_f32(a, b, acc);
  }
#pragma unroll
  for (int j = 0; j < 8; ++j) {
    int row = mBase + j + half * 8;
    C[(size_t)row * ldc + nBase + mn] = acc[j];
  }
}

// ------------------------------------------------ fused K-head attention -----
// s_u[u,k,d] = sum_i softmax_k(ue[u,k,:]·s_i[k,:,i]) * rel[u,i] * s_i[k,d,i]
// Block = 128 threads = 4 waves; wave w owns head w for a 16-user strip.
// Per 16-item tile: logits via WMMA -> raw logits to LDS (C-layout) -> barrier
// -> per-element 4-head softmax * rel -> own-head weight tile to LDS ->
// (same-wave readback, in-order) -> A-fragment from LDS, weighted-sum WMMAs.
__global__ void attention_kernel(const float* __restrict__ ue,      // [U,E]
                                 const float* __restrict__ si_dI2,  // pair layout
                                 const float* __restrict__ si_iD2,  // pair layout
                                 const float* __restrict__ rel,     // [U,I]
                                 float* __restrict__ s_u_out) {     // [U,K*D]
  __shared__ float lg[KH][16 * 16];
  __shared__ float wt[KH][16 * 16];
  const int tid = threadIdx.x;
  const int kh = tid >> 5;       // wave index == head index
  const int lane = tid & 31;
  const int half = lane >> 4;
  const int mn = lane & 15;
  const int ub = blockIdx.x * 16;

  // hoist this head's ue A-fragments (loop-invariant): 8 x v2f = 16 VGPRs
  v2f afrag[8];
#pragma unroll
  for (int kk = 0; kk < 8; ++kk) {
    int kd = kk * 4 + half * 2;
    afrag[kk] = *(const v2f*)(ue + (size_t)(ub + mn) * EDIM + kh * DH + kd);
  }

  v8f sacc0 = {0.f,0.f,0.f,0.f,0.f,0.f,0.f,0.f};
  v8f sacc1 = {0.f,0.f,0.f,0.f,0.f,0.f,0.f,0.f};
  const float* bDI = si_dI2 + (size_t)kh * DH * IDIM;  // this head's pair block
  const float* bID = si_iD2 + (size_t)kh * IDIM * DH;

  for (int it = 0; it < IDIM / 16; ++it) {
    const int ib = it * 16;
    // ---- logits tile for this head: K-dim = D = 32 (8 wmma steps)
    v8f c = {0.f,0.f,0.f,0.f,0.f,0.f,0.f,0.f};
#pragma unroll
    for (int kk = 0; kk < 8; ++kk) {
      int kd = kk * 4 + half * 2;
      v2f b = *(const v2f*)(bDI + ((size_t)(kd >> 1) * IDIM + ib + mn) * 2);
      c = wmma_f32(afrag[kk], b, c);
    }
    // ---- raw logits -> LDS (C-layout [m][n])
#pragma unroll
    for (int j = 0; j < 8; ++j) lg[kh][(j + half * 8) * 16 + mn] = c[j];
    __syncthreads();
    // ---- 4-head softmax * rel; each wave emits its own head's weight tile
#pragma unroll
    for (int j = 0; j < 8; ++j) {
      int m = j + half * 8;
      int li = m * 16 + mn;
      float l0 = lg[0][li], l1 = lg[1][li], l2 = lg[2][li], l3 = lg[3][li];
      float mx = fmaxf(fmaxf(l0, l1), fmaxf(l2, l3));
      float e0 = __expf(l0 - mx), e1 = __expf(l1 - mx);
      float e2 = __expf(l2 - mx), e3 = __expf(l3 - mx);
      float ek = (kh == 0) ? e0 : (kh == 1) ? e1 : (kh == 2) ? e2 : e3;
      float rv = rel[(size_t)(ub + m) * IDIM + ib + mn];
      wt[kh][li] = ek * rv / (e0 + e1 + e2 + e3);
    }
    // own-tile readback below is same-wave; LDS is in-order per wave -> no barrier
    // ---- weighted accumulate: sacc += W(16x16) * s_iT[ib..ib+15, 0..31]
#pragma unroll
    for (int kk = 0; kk < 4; ++kk) {
      int ki = kk * 4 + half * 2;
      v2f a = *(const v2f*)(&wt[kh][mn * 16 + ki]);
      v2f b0 = *(const v2f*)(bID + ((size_t)((ib + ki) >> 1) * DH + mn) * 2);
      v2f b1 = *(const v2f*)(bID + ((size_t)((ib + ki) >> 1) * DH + 16 + mn) * 2);
      sacc0 = wmma_f32(a, b0, sacc0);
      sacc1 = wmma_f32(a, b1, sacc1);
    }
    __syncthreads();  // protect lg before next tile overwrites it
  }
  // ---- store this head's s_u slice [U, K*D]
#pragma unroll
  for (int j = 0; j < 8; ++j) {
    int row = ub + j + half * 8;
    s_u_out[(size_t)row * (KH * DH) + kh * DH + mn] = sacc0[j];
    s_u_out[(size_t)row * (KH * DH) + kh * DH + 16 + mn] = sacc1[j];
  }
}

// tun = l2norm(s_u0 + s_u) over d; user_feature[u, k*D+d] += ubw[u,r] * tun
__global__ void tun_kernel(const float* __restrict__ s_u0, const float* __restrict__ s_u,
                           const float* __restrict__ ubw, float* __restrict__ user_feature,
                           int rIdx) {
  int bid = blockIdx.x;
  int k = bid & 3;
  int u = bid >> 2;
  int d = threadIdx.x;
  size_t off = (size_t)u * KH * DH + k * DH + d;
  float v = s_u0[off] + s_u[off];
  float sq = v * v;
#pragma unroll
  for (int o = 16; o > 0; o >>= 1) sq += __shfl_xor(sq, o, 32);
  float t = v / fmaxf(sqrtf(sq), 1e-12f);
  user_feature[(size_t)u * EDIM + k * DH + d] += ubw[u * RREL + rIdx] * t;
}

// score2[b,l] += dot(proj[user[b]], tip[item[b,l]])
__global__ void score2_kernel(const int* __restrict__ user, const int* __restrict__ item,
                              const float* __restrict__ proj, const float* __restrict__ tip,
                              float* __restrict__ score2) {
  int idx = blockIdx.x * blockDim.x + threadIdx.x;
  if (idx >= BB * LLEN) return;
  int b = idx / LLEN;
  int iu = user[b];
  int ii = item[idx];
  const float4* up = (const float4*)(proj + (size_t)iu * EDIM);
  const float4* ip = (const float4*)(tip + (size_t)ii * EDIM);
  float s = 0.f;
#pragma unroll 8
  for (int e = 0; e < EDIM / 4; ++e) {
    float4 a = up[e], c = ip[e];
    s += a.x * c.x + a.y * c.y + a.z * c.z + a.w * c.w;
  }
  score2[idx] += s;
}

// scores = score1 + LAMB*score2/R ; per-block partial of L2 pieces -> partials
__global__ void score_kernel(const int* __restrict__ user, const int* __restrict__ item,
                             const float* __restrict__ ufW, const float* __restrict__ ifeat,
                             const float* __restrict__ score2, float* __restrict__ scores,
                             float* __restrict__ partials) {
  __shared__ float sdata[128];
  int tid = threadIdx.x;
  int idx = blockIdx.x * 128 + tid;
  float l2p = 0.f;
  if (idx < BB * LLEN) {
    int b = idx / LLEN;
    int l = idx % LLEN;
    int iu = user[b];
    int ii = item[idx];
    const float4* up = (const float4*)(ufW + (size_t)iu * EDIM);
    const float4* ip = (const float4*)(ifeat + (size_t)ii * EDIM);
    float s1 = 0.f, itsq = 0.f, usq = 0.f;
#pragma unroll 8
    for (int e = 0; e < EDIM / 4; ++e) {
      float4 a = up[e], c = ip[e];
      s1 += a.x * c.x + a.y * c.y + a.z * c.z + a.w * c.w;
      itsq += c.x * c.x + c.y * c.y + c.z * c.z + c.w * c.w;
      usq += a.x * a.x + a.y * a.y + a.z * a.z + a.w * a.w;
    }
    scores[idx] = s1 + (LAMB_C / (float)RREL) * score2[idx];
    l2p = itsq + ((l == 0) ? (float)LLEN * usq : 0.f);
  }
  sdata[tid] = l2p;
  __syncthreads();
#pragma unroll
  for (int o = 64; o > 0; o >>= 1) {
    if (tid < o) sdata[tid] += sdata[tid + o];
    __syncthreads();
  }
  if (tid == 0) partials[blockIdx.x] = sdata[0];
}

__global__ void l2_reduce_kernel(const float* __restrict__ partials, int n,
                                 float* __restrict__ out) {
  __shared__ float sd[256];
  int tid = threadIdx.x;
  float s = 0.f;
  for (int i = tid; i < n; i += 256) s += partials[i];
  sd[tid] = s;
  __syncthreads();
#pragma unroll
  for (int o = 128; o > 0; o >>= 1) {
    if (tid < o) sd[tid] += sd[tid + o];
    __syncthreads();
  }
  if (tid == 0) out[0] = L2NORM_C * sd[0];
}

// ----------------------------------------------------------------- launch ----
extern "C" void kernel_launch(void* const* d_in, const int* in_sizes, int n_in,
                              void* d_out, int out_size, void* d_ws, size_t ws_size,
                              hipStream_t stream) {
  const int* user = (const int*)d_in[0];
  const int* item = (const int*)d_in[1];
  const float* user_embedding = (const float*)d_in[2];
  const float* item_embedding = (const float*)d_in[3];
  const float* mgnn_weight = (const float*)d_in[4];
  const float* item_behaviour_W = (const float*)d_in[5];
  const float* item_propagate_W = (const float*)d_in[6];
  const float* W = (const float*)d_in[7];
  const float* W_K = (const float*)d_in[8];
  const float* B_K = (const float*)d_in[9];
  const float* train_matrix = (const float*)d_in[10];
  const float* relation_mats = (const float*)d_in[11];
  const float* item_graphs = (const float*)d_in[12];
  const float* item_graph_degree = (const float*)d_in[13];
  const float* user_behaviour_degree = (const float*)d_in[14];

  float* scores = (float*)d_out;             // [B,L]
  float* l2_out = scores + BB * LLEN;        // scalar

  // workspace carve-out (floats)
  float* ws = (float*)d_ws;
  const size_t KDI = (size_t)KH * DH * IDIM;   // 524288
  const size_t MAT = (size_t)IDIM * EDIM;      // 524288
  float* si_dI2 = ws;               ws += KDI;  // pair layout (logits B)
  float* si_iD2 = ws;               ws += KDI;  // pair layout (weighted-sum B)
  float* s_u0   = ws;               ws += KDI;  // [U,K,D]
  float* s_u    = ws;               ws += KDI;  // [U,K,D]
  float* tipp   = ws;               ws += MAT;  // [I,E] pre / scaled
  float* tip    = ws;               ws += MAT;  // [I,E]
  float* tuinp  = ws;               ws += MAT;  // [U,E]
  float* proj   = ws;               ws += MAT;  // [U,E]
  float* ufeat  = ws;               ws += MAT;  // [U,E]
  float* ufW    = ws;               ws += MAT;  // [U,E]
  float* ifpre  = ws;               ws += MAT;  // [I,E]
  float* ifeat  = ws;               ws += MAT;  // [I,E]
  float* ubw    = ws;               ws += (size_t)UDIM * RREL;
  float* score2 = ws;               ws += (size_t)BB * LLEN;
  float* partials = ws;             ws += 512;

  // init accumulators
  zero_kernel<<<(UDIM * EDIM + 255) / 256, 256, 0, stream>>>(ufeat, UDIM * EDIM);
  zero_kernel<<<(BB * LLEN + 255) / 256, 256, 0, stream>>>(score2, BB * LLEN);

  // behaviour weighting
  ubw_kernel<<<UDIM / 256, 256, 0, stream>>>(user_behaviour_degree, mgnn_weight, ubw);

  // K-head projections (items: two pair layouts; users: [U,K,D])
  head_proj_kernel<<<IDIM * KH, 32, 0, stream>>>(item_embedding, W_K, B_K, IDIM,
                                                 si_dI2, si_iD2, nullptr);
  head_proj_kernel<<<UDIM * KH, 32, 0, stream>>>(user_embedding, W_K, B_K, UDIM,
                                                 nullptr, nullptr, s_u0);

  // item_feature = (train_matrix.T @ user_embedding) @ W
  wmma_gemm_f32<true><<<dim3(EDIM / 16, IDIM / 16), 32, 0, stream>>>(
      train_matrix, user_embedding, ifpre, IDIM, EDIM, UDIM, IDIM, EDIM, EDIM);
  wmma_gemm_f32<false><<<dim3(EDIM / 16, IDIM / 16), 32, 0, stream>>>(
      ifpre, W, ifeat, IDIM, EDIM, EDIM, EDIM, EDIM, EDIM);

  for (int r = 0; r < RREL; ++r) {
    const float* rel = relation_mats + (size_t)r * UDIM * IDIM;
    const float* ig = item_graphs + (size_t)r * IDIM * IDIM;

    // tip = ((item_graphs[r] @ item_emb) / (deg+EPS)) @ item_propagate_W[r]
    wmma_gemm_f32<false><<<dim3(EDIM / 16, IDIM / 16), 32, 0, stream>>>(
        ig, item_embedding, tipp, IDIM, EDIM, IDIM, IDIM, EDIM, EDIM);
    row_scale_kernel<<<(IDIM * EDIM + 255) / 256, 256, 0, stream>>>(
        tipp, item_graph_degree + (size_t)r * IDIM, 1, IDIM * EDIM, EDIM);
    wmma_gemm_f32<false><<<dim3(EDIM / 16, IDIM / 16), 32, 0, stream>>>(
        tipp, item_propagate_W + (size_t)r * EDIM * EDIM, tip, IDIM, EDIM, EDIM,
        EDIM, EDIM, EDIM);

    // fused attention -> s_u ; then tun + user_feature accumulation
    attention_kernel<<<UDIM / 16, 128, 0, stream>>>(user_embedding, si_dI2, si_iD2,
                                                    rel, s_u);
    tun_kernel<<<UDIM * KH, 32, 0, stream>>>(s_u0, s_u, ubw, ufeat, r);

    // tuinp = (rel @ tip) / (ubd[:,r]+EPS) ; proj = tuinp @ item_behaviour_W[r]
    wmma_gemm_f32<false><<<dim3(EDIM / 16, UDIM / 16), 32, 0, stream>>>(
        rel, tip, tuinp, UDIM, EDIM, IDIM, IDIM, EDIM, EDIM);
    row_scale_kernel<<<(UDIM * EDIM + 255) / 256, 256, 0, stream>>>(
        tuinp, user_behaviour_degree + r, RREL, UDIM * EDIM, EDIM);
    wmma_gemm_f32<false><<<dim3(EDIM / 16, UDIM / 16), 32, 0, stream>>>(
        tuinp, item_behaviour_W + (size_t)r * EDIM * EDIM, proj, UDIM, EDIM, EDIM,
        EDIM, EDIM, EDIM);

    // score2 += <proj[user[b]], tip[item[b,l]]>
    score2_kernel<<<(BB * LLEN + 255) / 256, 256, 0, stream>>>(user, item, proj, tip,
                                                               score2);
  }

  // user_feature @ W
  wmma_gemm_f32<false><<<dim3(EDIM / 16, UDIM / 16), 32, 0, stream>>>(
      ufeat, W, ufW, UDIM, EDIM, EDIM, EDIM, EDIM, EDIM);

  // final scores + L2
  const int nScoreBlocks = (BB * LLEN + 127) / 128;  // 400
  score_kernel<<<nScoreBlocks, 128, 0, stream>>>(user, item, ufW, ifeat, score2,
                                                 scores, partials);
  l2_reduce_kernel<<<1, 256, 0, stream>>>(partials, nScoreBlocks, l2_out);
}